// FuseRec_9096740733367
// MI455X (gfx1250) — compile-verified
//
#include <hip/hip_runtime.h>
#include <math.h>
#include <stdint.h>

// ---------------------------------------------------------------------------
// FuseRec forward for MI455X (gfx1250).
//  * All GEMMs: V_WMMA_F32_16X16X4_F32 (wave32, exact f32).
//  * l1 folded into W_ih (linear x linear), removing the B*L*64 intermediate.
//  * LSTM weights staged to LDS via the Tensor Data Mover (TENSOR_LOAD_TO_LDS,
//    TENSORcnt), per-step feature gathers via GLOBAL_LOAD_ASYNC_TO_LDS_B32
//    (ASYNCcnt) -- the gfx1250 async data paths.
// ---------------------------------------------------------------------------

typedef __attribute__((ext_vector_type(2))) float v2f;
typedef __attribute__((ext_vector_type(8))) float v8f;
typedef __attribute__((ext_vector_type(4))) unsigned int v4u;
typedef __attribute__((ext_vector_type(8))) int v8i_;
typedef __attribute__((ext_vector_type(4))) int v4i_;

constexpr int D    = 64;
constexpr int C_   = 20;
constexpr int Bb   = 2048;
constexpr int L    = 50;
constexpr int Kf   = 10;
constexpr int L2_  = 30;
constexpr int FEAT = D + C_;   // 84
constexpr int G4   = 4 * D;    // 256

__device__ __forceinline__ float sigf(float x) { return 1.0f / (1.0f + expf(-x)); }
__device__ __forceinline__ float lrelu(float x) { return x < 0.0f ? 0.01f * x : x; }

// ISA 10.2: for a generic address inside the LDS aperture, addr[31:0] IS the
// LDS byte offset (upper bits are the aperture tag) -> truncation is the
// correct way to form VDST / D#.lds_addr values.
__device__ __forceinline__ uint32_t lds_off(const void* p) {
  return (uint32_t)(uintptr_t)p;
}

// D = A(16x4 f32) * B(4x16 f32) + C(16x16 f32), one WMMA per wave32.
__device__ __forceinline__ v8f wmma_f32(v2f a, v2f b, v8f c) {
  return __builtin_amdgcn_wmma_f32_16x16x4_f32(
      /*neg_a=*/false, a, /*neg_b=*/false, b,
      /*c_mod=*/(short)0, c, /*reuse_a=*/false, /*reuse_b=*/false);
}

// Fragment loader for a row-major [16 x stride] LDS tile (CDNA5 16x4 A and
// 4x16 B lane layouts are symmetric): lanes 0-15 hold K = k4+0/+1,
// lanes 16-31 hold K = k4+2/+3.
__device__ __forceinline__ v2f load_frag(const float* s, int stride, int k4, int lane) {
  int m  = lane & 15;
  int k0 = k4 + ((lane >> 4) << 1);
  v2f r;
  r.x = s[m * stride + k0];
  r.y = s[m * stride + k0 + 1];
  return r;
}

// ---------------------------------------------------------------------------
// Tensor Data Mover: 1-D tile (n f32 elements, n <= 65535) global -> LDS.
// D# per ISA 8.3/8.4: group0 = {count=1 | lds_addr | global_addr | type=2},
// group1 = {data_size=4B, tensor_dim0=n, tensor_dim1=1, tile_dim0=n,
// tile_dim1=1, dim0_stride=n}; groups 2/3 (and the extra group of the 6-arg
// builtin) zero => dims unused (<=2D).  Completion via S_WAIT_TENSORCNT.
// ---------------------------------------------------------------------------
#if __has_builtin(__builtin_amdgcn_tensor_load_to_lds)
#define HAVE_TDM 1
__device__ __forceinline__ void tdm_load_1d(uint32_t lds_addr, const void* gptr,
                                            uint32_t n) {
  uint64_t ga = (uint64_t)(uintptr_t)gptr;
  v4u g0;
  g0.x = 1u;                                          // count=1
  g0.y = lds_addr;                                    // D#.lds_addr
  g0.z = (uint32_t)(ga & 0xFFFFFFFFu);                // global_addr[31:0]
  g0.w = (uint32_t)((ga >> 32) & 0x01FFFFFFu) | (2u << 30);  // [56:32] | type=2
  v8i_ g1;
  g1[0] = (int)(2u << 16);                            // data_size = 4 bytes
  g1[1] = (int)((n & 0xFFFFu) << 16);                 // tensor_dim0 lo
  g1[2] = (int)((n >> 16) | (1u << 16));              // tensor_dim0 hi | dim1=1
  g1[3] = (int)((n & 0xFFFFu) << 16);                 // tile_dim0 = n
  g1[4] = 1;                                          // tile_dim1 = 1
  g1[5] = (int)n;                                     // tensor_dim0_stride = n
  g1[6] = 0;
  g1[7] = 0;
  v4i_ gz = {0, 0, 0, 0};
  v8i_ gz8 = {0, 0, 0, 0, 0, 0, 0, 0};
  __builtin_amdgcn_tensor_load_to_lds(g0, g1, gz, gz, gz8, 0);
}
#else
#define HAVE_TDM 0
#endif

__device__ __forceinline__ void wait_tensorcnt0() {
#if __has_builtin(__builtin_amdgcn_s_wait_tensorcnt)
  __builtin_amdgcn_s_wait_tensorcnt((short)0);
#else
  asm volatile("s_wait_tensorcnt 0x0" ::: "memory");
#endif
}

__device__ __forceinline__ void wait_asynccnt0() {
#if __has_builtin(__builtin_amdgcn_s_wait_asynccnt)
  __builtin_amdgcn_s_wait_asynccnt((short)0);
#else
  asm volatile("s_wait_asynccnt 0x0" ::: "memory");
#endif
}

// GLOBAL_LOAD_ASYNC_TO_LDS_B32, GV mode: per-lane 64-bit global address,
// per-lane 32-bit LDS destination. Tracked with ASYNCcnt.
__device__ __forceinline__ void async_gather_b32(uint32_t lds_addr, const float* src) {
  asm volatile("global_load_async_to_lds_b32 %0, %1, off"
               :
               : "v"(lds_addr), "v"(src)
               : "memory");
}

// ---------------------------------------------------------------------------
// Prep: W_eff[g][c] = sum_j W_ih[g][j] * l1_w[j][c]   (256 x 84)
//       bias_c[g]  = b_ih[g] + b_hh[g] + sum_j W_ih[g][j] * l1_b[j]
// ---------------------------------------------------------------------------
__global__ __launch_bounds__(128) void k_prep(
    const float* __restrict__ W_ih, const float* __restrict__ l1_w,
    const float* __restrict__ l1_b, const float* __restrict__ b_ih,
    const float* __restrict__ b_hh, float* __restrict__ W_eff,
    float* __restrict__ bias_c) {
  int g = blockIdx.x;          // 0..255
  int c = threadIdx.x;
  if (c < FEAT) {
    float s = 0.0f;
    for (int j = 0; j < D; ++j) s += W_ih[g * D + j] * l1_w[j * FEAT + c];
    W_eff[g * FEAT + c] = s;
  } else if (c == FEAT) {
    float s = b_ih[g] + b_hh[g];
    for (int j = 0; j < D; ++j) s += W_ih[g * D + j] * l1_b[j];
    bias_c[g] = s;
  }
}

// ---------------------------------------------------------------------------
// LSTM over 50 steps.  One block = 16 batch rows, 128 threads = 4 waves.
// Wave w owns gate columns [w*64, w*64+64) (gate w of i/f/g/o) as 4 WMMA
// N-tiles.  Weights resident in 182 KB dynamic LDS, staged by the TDM.
// ---------------------------------------------------------------------------
__global__ __launch_bounds__(128) void k_lstm(
    const int* __restrict__ u_items, const int* __restrict__ u_items_mask,
    const float* __restrict__ item_emb, const float* __restrict__ i_class,
    const float* __restrict__ W_eff, const float* __restrict__ W_hh,
    const float* __restrict__ bias_c, float* __restrict__ hu) {
  extern __shared__ float smem[];
  float* sW1 = smem;                    // 256*84
  float* sWh = sW1 + G4 * FEAT;         // 256*64
  float* sB  = sWh + G4 * D;            // 256
  float* sF  = sB + G4;                 // 16*84
  float* sH  = sF + 16 * FEAT;          // 16*64
  float* sC  = sH + 16 * D;             // 16*64
  float* sG  = sC + 16 * D;             // 16*256
  __shared__ int smask[16];

  const int tid = threadIdx.x;
  const int r0  = blockIdx.x * 16;

#if HAVE_TDM
  if (tid < 32) {  // wave 0 issues both weight DMAs (wave-uniform branch)
    tdm_load_1d(lds_off(sW1), W_eff, G4 * FEAT);
    tdm_load_1d(lds_off(sWh), W_hh, G4 * D);
    wait_tensorcnt0();
  }
#else
  for (int e = tid; e < G4 * FEAT; e += 128) sW1[e] = W_eff[e];
  for (int e = tid; e < G4 * D; e += 128) sWh[e] = W_hh[e];
#endif
  for (int e = tid; e < G4; e += 128) sB[e] = bias_c[e];
  for (int e = tid; e < 16 * D; e += 128) { sH[e] = 0.0f; sC[e] = 0.0f; }
  if (tid < 16) smask[tid] = u_items_mask[r0 + tid];
  __syncthreads();

  const int lane = tid & 31;
  const int wv   = tid >> 5;
  const int nl   = lane & 15;
  const int mh   = (lane >> 4) * 8;
  const uint32_t sF_base = lds_off(sF);

  for (int t = 0; t < L; ++t) {
    // Gather x_t features [item_emb[idx] | i_class[idx]] straight into LDS
    // via async-to-LDS loads (per-lane scattered addresses, ASYNCcnt).
    for (int e = tid; e < 16 * FEAT; e += 128) {
      int m = e / FEAT, c = e - m * FEAT;
      int idx = u_items[(r0 + m) * L + t];
      const float* src = (c < D) ? (item_emb + (size_t)idx * D + c)
                                 : (i_class + (size_t)idx * C_ + (c - D));
      async_gather_b32(sF_base + (uint32_t)e * 4u, src);
    }
    wait_asynccnt0();
    __syncthreads();

    v8f acc[4] = {v8f{}, v8f{}, v8f{}, v8f{}};
    // g += feat @ W_eff^T (K=84): batch all 5 fragment loads of a k-step
    // before the 4 WMMAs so DS waits can be partial / overlapped.
#pragma unroll
    for (int k = 0; k < FEAT / 4; ++k) {
      v2f a  = load_frag(sF, FEAT, k * 4, lane);
      v2f b0 = load_frag(sW1 + (wv * 64 + 0)  * FEAT, FEAT, k * 4, lane);
      v2f b1 = load_frag(sW1 + (wv * 64 + 16) * FEAT, FEAT, k * 4, lane);
      v2f b2 = load_frag(sW1 + (wv * 64 + 32) * FEAT, FEAT, k * 4, lane);
      v2f b3 = load_frag(sW1 + (wv * 64 + 48) * FEAT, FEAT, k * 4, lane);
      acc[0] = wmma_f32(a, b0, acc[0]);
      acc[1] = wmma_f32(a, b1, acc[1]);
      acc[2] = wmma_f32(a, b2, acc[2]);
      acc[3] = wmma_f32(a, b3, acc[3]);
    }
    // g += h @ W_hh^T (K=64)
#pragma unroll
    for (int k = 0; k < D / 4; ++k) {
      v2f a  = load_frag(sH, D, k * 4, lane);
      v2f b0 = load_frag(sWh + (wv * 64 + 0)  * D, D, k * 4, lane);
      v2f b1 = load_frag(sWh + (wv * 64 + 16) * D, D, k * 4, lane);
      v2f b2 = load_frag(sWh + (wv * 64 + 32) * D, D, k * 4, lane);
      v2f b3 = load_frag(sWh + (wv * 64 + 48) * D, D, k * 4, lane);
      acc[0] = wmma_f32(a, b0, acc[0]);
      acc[1] = wmma_f32(a, b1, acc[1]);
      acc[2] = wmma_f32(a, b2, acc[2]);
      acc[3] = wmma_f32(a, b3, acc[3]);
    }
    // Spill gate pre-activations (+bias) to LDS.
#pragma unroll
    for (int nt = 0; nt < 4; ++nt) {
      int n = wv * 64 + nt * 16 + nl;
      float bn = sB[n];
#pragma unroll
      for (int r = 0; r < 8; ++r) sG[(r + mh) * G4 + n] = acc[nt][r] + bn;
    }
    __syncthreads();

    // LSTM cell update: cols [0,64)=i, [64,128)=f, [128,192)=g, [192,256)=o.
    for (int e = tid; e < 16 * D; e += 128) {
      int m = e >> 6, d = e & 63;
      float gi = sigf(sG[m * G4 + d]);
      float gf = sigf(sG[m * G4 + 64 + d]);
      float gg = tanhf(sG[m * G4 + 128 + d]);
      float go = sigf(sG[m * G4 + 192 + d]);
      float cc = gf * sC[e] + gi * gg;
      float hh = go * tanhf(cc);
      sC[e] = cc;
      sH[e] = hh;
      if (t == smask[m] - 1) hu[(size_t)(r0 + m) * D + d] = hh;  // out[b, mask-1]
    }
    __syncthreads();
  }
}

// ---------------------------------------------------------------------------
// Generic WMMA GEMM:  C[M x 64] = rs[m] * (A[M x KK] @ W[64 x KK]^T + bs*bias)
// Block = 128 threads (4 waves), 16 rows per block, one 16x16 N-tile / wave.
// KK is compile-time -> fully unrolled k-loop with immediate DS offsets.
// ---------------------------------------------------------------------------
template <int KK>
__global__ __launch_bounds__(128) void k_gemm_nt(
    const float* __restrict__ A, const float* __restrict__ W,
    const float* __restrict__ bias, float bias_scale,
    const float* __restrict__ row_scale, float* __restrict__ Cout) {
  __shared__ float sA[16 * KK];
  __shared__ float sW[64 * KK];
  const int tid = threadIdx.x;
  const int r0  = blockIdx.x * 16;

  for (int e = tid; e < 64 * KK; e += 128) sW[e] = W[e];
  for (int e = tid; e < 16 * KK; e += 128) sA[e] = A[(size_t)r0 * KK + e];
  __syncthreads();

  const int lane = tid & 31;
  const int wv   = tid >> 5;
  const float* sWn = sW + (wv * 16) * KK;

  v8f acc = {};
#pragma unroll
  for (int k = 0; k < KK / 4; ++k) {
    v2f a = load_frag(sA, KK, k * 4, lane);
    v2f b = load_frag(sWn, KK, k * 4, lane);
    acc = wmma_f32(a, b, acc);
  }

  int n  = wv * 16 + (lane & 15);
  int mh = (lane >> 4) * 8;
  float bn = bias_scale * bias[n];
#pragma unroll
  for (int r = 0; r < 8; ++r) {
    int m = r + mh;
    float v = acc[r] + bn;
    if (row_scale) v *= row_scale[r0 + m];
    Cout[(size_t)(r0 + m) * 64 + n] = v;
  }
}

// ---------------------------------------------------------------------------
// sumfeat[row=(b,k)] = sum_{l<30} [item_emb | i_class][u_frids_items[b,k,l]]
// rowscale[row] = 1 / u_frids_mask[b]
// ---------------------------------------------------------------------------
__global__ __launch_bounds__(128) void k_sumfeat(
    const int* __restrict__ u_frids_items, const int* __restrict__ u_frids_mask,
    const float* __restrict__ item_emb, const float* __restrict__ i_class,
    float* __restrict__ sumfeat, float* __restrict__ rowscale) {
  int row = blockIdx.x;  // b*Kf + k
  int tid = threadIdx.x;
  __shared__ int sidx[L2_];
  if (tid < L2_) sidx[tid] = u_frids_items[(size_t)row * L2_ + tid];
  __syncthreads();
  if (tid < FEAT) {
    float s = 0.0f;
    for (int l = 0; l < L2_; ++l) {
      int idx = sidx[l];
      s += (tid < D) ? item_emb[(size_t)idx * D + tid]
                     : i_class[(size_t)idx * C_ + (tid - D)];
    }
    sumfeat[(size_t)row * FEAT + tid] = s;
  }
  if (tid == FEAT) rowscale[row] = 1.0f / (float)u_frids_mask[row / Kf];
}

// ---------------------------------------------------------------------------
// Friend attention:  at_k = lrelu([u,v_k] . l4_w + l4_b), masked softmax over
// k < mask[b], su[b] = sum_k auv_k * pv[b,k,:]
// ---------------------------------------------------------------------------
__global__ __launch_bounds__(64) void k_att(
    const int* __restrict__ users, const int* __restrict__ u_frids,
    const int* __restrict__ u_frids_mask, const float* __restrict__ user_emb,
    const float* __restrict__ l4_w, const float* __restrict__ l4_b,
    const float* __restrict__ pv, float* __restrict__ su) {
  int b = blockIdx.x, tid = threadIdx.x;
  __shared__ float s_at[Kf];
  __shared__ float s_auv[Kf];
  if (tid < Kf) {
    int uu = users[b];
    int fr = u_frids[b * Kf + tid];
    float dot = l4_b[0];
    for (int d = 0; d < D; ++d)
      dot += l4_w[d] * user_emb[(size_t)uu * D + d] +
             l4_w[D + d] * user_emb[(size_t)fr * D + d];
    s_at[tid] = lrelu(dot);
  }
  __syncthreads();
  if (tid == 0) {
    int mk = u_frids_mask[b];  // >= 1
    float mx = -3.4e38f;
    for (int k = 0; k < mk; ++k) mx = fmaxf(mx, s_at[k]);
    float s = 0.0f;
    for (int k = 0; k < mk; ++k) s += expf(s_at[k] - mx);
    for (int k = 0; k < Kf; ++k)
      s_auv[k] = (k < mk) ? expf(s_at[k] - mx) / s : 0.0f;
  }
  __syncthreads();
  float accv = 0.0f;
  for (int k = 0; k < Kf; ++k) accv += s_auv[k] * pv[((size_t)b * Kf + k) * 64 + tid];
  su[(size_t)b * 64 + tid] = accv;
}

// out[b,0:192] = [x, i, x*i]  with i = item_emb[items[b]]
__global__ void k_concat(const float* __restrict__ x, const int* __restrict__ items,
                         const float* __restrict__ item_emb, float* __restrict__ out) {
  int e = blockIdx.x * blockDim.x + threadIdx.x;
  if (e >= Bb * 192) return;
  int b = e / 192, c = e - b * 192;
  int ii = items[b];
  float v;
  if (c < 64)        v = x[(size_t)b * 64 + c];
  else if (c < 128)  v = item_emb[(size_t)ii * 64 + (c - 64)];
  else               v = x[(size_t)b * 64 + (c - 128)] * item_emb[(size_t)ii * 64 + (c - 128)];
  out[e] = v;
}

// z[j][b][k] = lrelu([i, item_emb[F_i[b,j,k]]] . l6_w + l6_b)
__global__ __launch_bounds__(32) void k_z(
    const int* __restrict__ items, const int* __restrict__ F_i,
    const float* __restrict__ item_emb, const float* __restrict__ l6_w,
    const float* __restrict__ l6_b, float* __restrict__ z) {
  int b = blockIdx.x, k = threadIdx.x;
  if (k >= Kf) return;
  int ii = items[b];
  int f1 = F_i[(b * 2 + 0) * Kf + k];
  int f2 = F_i[(b * 2 + 1) * Kf + k];
  float d1 = l6_b[0], d2 = l6_b[0];
  for (int d = 0; d < D; ++d) {
    float iv = item_emb[(size_t)ii * D + d];
    d1 += l6_w[d] * iv + l6_w[D + d] * item_emb[(size_t)f1 * D + d];
    d2 += l6_w[d] * iv + l6_w[D + d] * item_emb[(size_t)f2 * D + d];
  }
  z[(size_t)b * Kf + k]           = lrelu(d1);
  z[(size_t)Bb * Kf + b * Kf + k] = lrelu(d2);
}

// Per-column (axis=0, across B) max & exp-sum for the two F_i softmaxes.
__global__ __launch_bounds__(256) void k_colreduce(
    const float* __restrict__ z, float* __restrict__ colmax, float* __restrict__ colsum) {
  int col = blockIdx.x;  // 0..19 : [0,10)=z1 cols, [10,20)=z2 cols
  int j = col / Kf, k = col % Kf;
  const float* zp = z + (size_t)j * Bb * Kf + k;
  __shared__ float red[256];
  int tid = threadIdx.x;
  float mx = -3.4e38f;
  for (int b = tid; b < Bb; b += 256) mx = fmaxf(mx, zp[(size_t)b * Kf]);
  red[tid] = mx;
  __syncthreads();
  for (int s = 128; s > 0; s >>= 1) {
    if (tid < s) red[tid] = fmaxf(red[tid], red[tid + s]);
    __syncthreads();
  }
  mx = red[0];
  __syncthreads();
  float sm = 0.0f;
  for (int b = tid; b < Bb; b += 256) sm += expf(zp[(size_t)b * Kf] - mx);
  red[tid] = sm;
  __syncthreads();
  for (int s = 128; s > 0; s >>= 1) {
    if (tid < s) red[tid] += red[tid + s];
    __syncthreads();
  }
  if (tid == 0) { colmax[col] = mx; colsum[col] = red[0]; }
}

// yi + S1..S4 + sigmoid.  One block (64 threads) per batch row.
__global__ __launch_bounds__(64) void k_final(
    const float* __restrict__ z, const float* __restrict__ colmax,
    const float* __restrict__ colsum, const int* __restrict__ F_i,
    const float* __restrict__ item_emb, const float* __restrict__ hu,
    const float* __restrict__ hui, const float* __restrict__ su,
    const float* __restrict__ sui, const float* __restrict__ lambdas,
    const float* __restrict__ alpha, float* __restrict__ out) {
  int b = blockIdx.x, d = threadIdx.x;
  float a = alpha[0];
  float y = 0.0f;
  for (int k = 0; k < Kf; ++k) {
    float w1 = expf(z[(size_t)b * Kf + k] - colmax[k]) / colsum[k];
    float w2 = expf(z[(size_t)Bb * Kf + b * Kf + k] - colmax[Kf + k]) / colsum[Kf + k];
    int f1 = F_i[(b * 2 + 0) * Kf + k];
    int f2 = F_i[(b * 2 + 1) * Kf + k];
    y += a * w1 * item_emb[(size_t)f1 * D + d] +
         (1.0f - a) * w2 * item_emb[(size_t)f2 * D + d];
  }
  __shared__ float red[4][64];
  size_t o = (size_t)b * 64 + d;
  red[0][d] = hu[o] * y;
  red[1][d] = hui[o] * y;
  red[2][d] = su[o] * y;
  red[3][d] = sui[o] * y;
  __syncthreads();
  for (int s = 32; s > 0; s >>= 1) {
    if (d < s) {
#pragma unroll
      for (int q = 0; q < 4; ++q) red[q][d] += red[q][d + s];
    }
    __syncthreads();
  }
  if (d == 0) {
    float S = lambdas[0] * red[0][0] + lambdas[1] * red[1][0] +
              lambdas[2] * red[2][0] + lambdas[3] * red[3][0];
    out[b] = 1.0f / (1.0f + expf(-S));
  }
}

// ---------------------------------------------------------------------------
extern "C" void kernel_launch(void* const* d_in, const int* in_sizes, int n_in,
                              void* d_out, int out_size, void* d_ws, size_t ws_size,
                              hipStream_t stream) {
  const int*   users         = (const int*)d_in[0];
  const int*   items         = (const int*)d_in[1];
  // d_in[2] = labels (unused)
  const int*   u_items       = (const int*)d_in[3];
  const int*   u_items_mask  = (const int*)d_in[4];
  const int*   u_frids       = (const int*)d_in[5];
  const int*   u_frids_mask  = (const int*)d_in[6];
  const int*   u_frids_items = (const int*)d_in[7];
  const int*   F_i           = (const int*)d_in[8];
  // d_in[9] = flag (unused)
  const float* user_emb = (const float*)d_in[10];
  const float* item_emb = (const float*)d_in[11];
  const float* i_class  = (const float*)d_in[12];
  const float* W_ih     = (const float*)d_in[13];
  const float* W_hh     = (const float*)d_in[14];
  const float* b_ih     = (const float*)d_in[15];
  const float* b_hh     = (const float*)d_in[16];
  const float* l1_w     = (const float*)d_in[17];
  const float* l1_b     = (const float*)d_in[18];
  const float* l2_w     = (const float*)d_in[19];
  const float* l2_b     = (const float*)d_in[20];
  const float* l3_w     = (const float*)d_in[21];
  const float* l3_b     = (const float*)d_in[22];
  const float* l4_w     = (const float*)d_in[23];
  const float* l4_b     = (const float*)d_in[24];
  const float* l5_w     = (const float*)d_in[25];
  const float* l5_b     = (const float*)d_in[26];
  const float* l6_w     = (const float*)d_in[27];
  const float* l6_b     = (const float*)d_in[28];
  const float* lambdas  = (const float*)d_in[29];
  const float* alpha    = (const float*)d_in[30];
  float* out = (float*)d_out;

  // Workspace layout (floats).
  float* ws       = (float*)d_ws;
  float* W_eff    = ws;                          // 256*84   = 21504
  float* bias_c   = W_eff + G4 * FEAT;           // 256
  float* hu       = bias_c + G4;                 // 2048*64  = 131072
  float* concat   = hu + Bb * 64;                // 2048*192 = 393216
  float* hui      = concat + Bb * 192;           // 131072
  float* sumfeat  = hui + Bb * 64;               // 20480*84 = 1720320
  float* rowscale = sumfeat + Bb * Kf * FEAT;    // 20480
  float* pv       = rowscale + Bb * Kf;          // 20480*64 = 1310720
  float* su       = pv + Bb * Kf * 64;           // 131072
  float* sui      = su + Bb * 64;                // 131072
  float* z        = sui + Bb * 64;               // 2*20480
  float* colmax   = z + 2 * Bb * Kf;             // 20
  float* colsum   = colmax + 20;                 // 20

  // 1. Fold l1 into the LSTM input weights.
  k_prep<<<G4, 128, 0, stream>>>(W_ih, l1_w, l1_b, b_ih, b_hh, W_eff, bias_c);

  // 2. LSTM over 50 steps (weights TDM-staged into 182 KB dynamic LDS).
  const size_t lstm_lds =
      (size_t)(G4 * FEAT + G4 * D + G4 + 16 * FEAT + 16 * D + 16 * D + 16 * G4) *
      sizeof(float);
  k_lstm<<<Bb / 16, 128, lstm_lds, stream>>>(u_items, u_items_mask, item_emb,
                                             i_class, W_eff, W_hh, bias_c, hu);

  // 3. hui = [hu, i, hu*i] @ l2_w^T + l2_b
  k_concat<<<(Bb * 192 + 255) / 256, 256, 0, stream>>>(hu, items, item_emb, concat);
  k_gemm_nt<192><<<Bb / 16, 128, 0, stream>>>(concat, l2_w, l2_b, 1.0f, nullptr, hui);

  // 4. pv = (sumfeat @ l3_w^T + 30*l3_b) / mask[b]
  k_sumfeat<<<Bb * Kf, 128, 0, stream>>>(u_frids_items, u_frids_mask, item_emb,
                                         i_class, sumfeat, rowscale);
  k_gemm_nt<FEAT><<<Bb * Kf / 16, 128, 0, stream>>>(sumfeat, l3_w, l3_b,
                                                    (float)L2_, rowscale, pv);

  // 5. friend attention -> su, then sui = [su, i, su*i] @ l5_w^T + l5_b
  k_att<<<Bb, 64, 0, stream>>>(users, u_frids, u_frids_mask, user_emb, l4_w, l4_b,
                               pv, su);
  k_concat<<<(Bb * 192 + 255) / 256, 256, 0, stream>>>(su, items, item_emb, concat);
  k_gemm_nt<192><<<Bb / 16, 128, 0, stream>>>(concat, l5_w, l5_b, 1.0f, nullptr, sui);

  // 6. F_i batch-axis softmaxes -> yi -> S1..S4 -> sigmoid.
  k_z<<<Bb, 32, 0, stream>>>(items, F_i, item_emb, l6_w, l6_b, z);
  k_colreduce<<<2 * Kf, 256, 0, stream>>>(z, colmax, colsum);
  k_final<<<Bb, 64, 0, stream>>>(z, colmax, colsum, F_i, item_emb, hu, hui, su,
                                 sui, lambdas, alpha, out);
}